// Net18_37391985279004
// MI455X (gfx1250) — compile-verified
//
#include <hip/hip_runtime.h>
#include <hip/hip_bf16.h>

typedef __attribute__((ext_vector_type(2))) float v2f;
typedef __attribute__((ext_vector_type(8))) float v8f;

#define IN_F 18
#define HID  16
#define LDS_STRIDE 20   // 18 cols padded to 20 (zeros) so K=18 -> 5 WMMA steps of K=4

// ---------------- degree / normalization ----------------

__global__ void k_init_deg(float* __restrict__ deg, int n) {
    int i = blockIdx.x * blockDim.x + threadIdx.x;
    if (i < n) deg[i] = 1.0f;  // self-loop contribution
}

__global__ void k_deg_accum(const int* __restrict__ dst, float* __restrict__ deg, int e) {
    int i = blockIdx.x * blockDim.x + threadIdx.x;
    if (i < e) atomicAdd(&deg[dst[i]], 1.0f);
}

__global__ void k_deg_to_dis(float* __restrict__ deg, int n) {
    int i = blockIdx.x * blockDim.x + threadIdx.x;
    if (i < n) {
        float d = deg[i];
        deg[i] = (d > 0.0f) ? (1.0f / sqrtf(d)) : 0.0f;  // in place -> dis
    }
}

// ---------------- layer-1 GEMM via WMMA f32 16x16x4, LDS-staged A ----------------
// One wave computes a 16(row) x 16(col) tile of h = x @ W1.
// A 16x4 f32 layout: lane L holds M = L%16; VGPR v holds K = 2*(L/16) + v (per K-block).
// B 4x16 f32 layout (mirror): lane L holds N = L%16; VGPR v holds K = 2*(L/16) + v.
// C/D 16x16 f32: VGPR j, lane L -> element [j + 8*(L/16)][L%16].

__global__ void __launch_bounds__(256) k_gemm1_wmma(const float* __restrict__ x,
                                                    const float* __restrict__ W1,
                                                    float* __restrict__ h, int n) {
    __shared__ float smem[8][16 * LDS_STRIDE];   // per-wave private tile
    const int lane  = threadIdx.x & 31;
    const int wslot = threadIdx.x >> 5;
    const int wave  = blockIdx.x * (blockDim.x >> 5) + wslot;
    const int row0  = wave * 16;
    if (row0 >= n) return;                       // wave-uniform: EXEC stays all-1s
    float* tile = smem[wslot];

    const bool fullTile = (row0 + 16 <= n);      // wave-uniform

    // stage x[row0..row0+15][0..17] into LDS, coalesced; zero pad cols 18..19
    if (fullTile) {
        const float* xb = x + (size_t)row0 * IN_F;
#pragma unroll
        for (int it = 0; it < 9; ++it) {
            int idx = it * 32 + lane;            // 0..287 flat over 16x18
            int r   = idx / IN_F;
            int k   = idx - r * IN_F;
            tile[r * LDS_STRIDE + k] = xb[idx];
        }
    } else {
#pragma unroll
        for (int it = 0; it < 9; ++it) {
            int idx = it * 32 + lane;
            int r   = idx / IN_F;
            int k   = idx - r * IN_F;
            int gr  = row0 + r;
            tile[r * LDS_STRIDE + k] = (gr < n) ? x[(size_t)gr * IN_F + k] : 0.0f;
        }
    }
    tile[(lane >> 1) * LDS_STRIDE + 18 + (lane & 1)] = 0.0f;

    const int half = lane >> 4;
    const int ml   = lane & 15;

    v8f acc = {};
#pragma unroll
    for (int t = 0; t < 5; ++t) {
        const int kb = t * 4 + half * 2;
        // A operand: 8-byte LDS load; same-wave LDS ops are in-order
        v2f a = *(const v2f*)&tile[ml * LDS_STRIDE + kb];
        v2f b;
#pragma unroll
        for (int v = 0; v < 2; ++v) {
            const int k  = kb + v;
            const int kc = (k < IN_F) ? k : 0;             // clamp: load always valid
            float bw = W1[kc * HID + ml];
            b[v] = (k < IN_F) ? bw : 0.0f;                 // branchless select
        }
        acc = __builtin_amdgcn_wmma_f32_16x16x4_f32(
            /*neg_a=*/false, a, /*neg_b=*/false, b,
            /*c_mod=*/(short)0, acc, /*reuse_a=*/false, /*reuse_b=*/false);
    }

    // store D: lane writes rows row0+8*half+j, column ml (stride HID floats)
    if (fullTile) {
        float* p = h + (size_t)(row0 + 8 * half) * HID + ml;
#pragma unroll
        for (int j = 0; j < 8; ++j) p[j * HID] = acc[j];   // imm offsets, no guards
    } else {
#pragma unroll
        for (int j = 0; j < 8; ++j) {
            const int r = row0 + 8 * half + j;
            if (r < n) h[(size_t)r * HID + ml] = acc[j];
        }
    }
}

// ---------------- layer-1 aggregation ----------------

// self-loop init: agg1[i,:] = h[i,:] * dis[i]^2, float4 per thread
__global__ void k_init_agg1(const float* __restrict__ h, const float* __restrict__ dis,
                            float* __restrict__ agg1, int n) {
    int idx = blockIdx.x * blockDim.x + threadIdx.x;   // over n*4 float4 chunks
    if (idx < n * (HID / 4)) {
        int i   = idx >> 2;
        float w = dis[i];
        w *= w;
        float4 hv = ((const float4*)h)[idx];
        float4 o  = make_float4(hv.x * w, hv.y * w, hv.z * w, hv.w * w);
        ((float4*)agg1)[idx] = o;
    }
}

// one thread per (edge, 4-feature chunk): b128 gather + 4 f32 atomics
__global__ void k_scatter1(const int* __restrict__ src, const int* __restrict__ dst,
                           const float* __restrict__ dis, const float* __restrict__ h,
                           float* __restrict__ agg1, int e) {
    int t = blockIdx.x * blockDim.x + threadIdx.x;
    if (t < e * (HID / 4)) {
        int edge = t >> 2;
        int f4   = (t & 3) * 4;
        int s = src[edge];
        int d = dst[edge];
        float w = dis[s] * dis[d];
        float4 hv = *(const float4*)&h[(size_t)s * HID + f4];
        float* o  = &agg1[(size_t)d * HID + f4];
        atomicAdd(o + 0, hv.x * w);
        atomicAdd(o + 1, hv.y * w);
        atomicAdd(o + 2, hv.z * w);
        atomicAdd(o + 3, hv.w * w);
    }
}

// ---------------- relu + bias + layer-2 transform (HID -> 1) ----------------

__global__ void k_layer2(const float* __restrict__ agg1, const float* __restrict__ b1,
                         const float* __restrict__ W2, float* __restrict__ h2, int n) {
    int i = blockIdx.x * blockDim.x + threadIdx.x;
    if (i < n) {
        const float4* a4 = (const float4*)(agg1 + (size_t)i * HID);
        float acc = 0.0f;
#pragma unroll
        for (int q = 0; q < 4; ++q) {
            float4 a = a4[q];
            acc += fmaxf(a.x + b1[q * 4 + 0], 0.0f) * W2[q * 4 + 0];
            acc += fmaxf(a.y + b1[q * 4 + 1], 0.0f) * W2[q * 4 + 1];
            acc += fmaxf(a.z + b1[q * 4 + 2], 0.0f) * W2[q * 4 + 2];
            acc += fmaxf(a.w + b1[q * 4 + 3], 0.0f) * W2[q * 4 + 3];
        }
        h2[i] = acc;
    }
}

// ---------------- layer-2 aggregation ----------------

__global__ void k_init_out(const float* __restrict__ h2, const float* __restrict__ dis,
                           const float* __restrict__ b2, float* __restrict__ out, int n) {
    int i = blockIdx.x * blockDim.x + threadIdx.x;
    if (i < n) {
        float w = dis[i];
        out[i] = h2[i] * w * w + b2[0];
    }
}

__global__ void k_scatter2(const int* __restrict__ src, const int* __restrict__ dst,
                           const float* __restrict__ dis, const float* __restrict__ h2,
                           float* __restrict__ out, int e) {
    int i = blockIdx.x * blockDim.x + threadIdx.x;
    if (i < e) {
        int s = src[i];
        int d = dst[i];
        atomicAdd(&out[d], h2[s] * dis[s] * dis[d]);
    }
}

extern "C" void kernel_launch(void* const* d_in, const int* in_sizes, int n_in,
                              void* d_out, int out_size, void* d_ws, size_t ws_size,
                              hipStream_t stream) {
    const float* x  = (const float*)d_in[0];
    const int*   ei = (const int*)d_in[1];
    const float* W1 = (const float*)d_in[2];
    const float* b1 = (const float*)d_in[3];
    const float* W2 = (const float*)d_in[4];
    const float* b2 = (const float*)d_in[5];
    float* out = (float*)d_out;

    const int n = in_sizes[0] / IN_F;
    const int e = in_sizes[1] / 2;
    const int* src = ei;
    const int* dst = ei + e;

    // workspace layout (floats), 16B-aligned regions:
    //   dis[n] | h[16n] | agg1[16n]; h2[n] reuses h region (h dead after scatter1)
    const size_t nA = ((size_t)n + 3) & ~(size_t)3;
    float* ws   = (float*)d_ws;
    float* dis  = ws;
    float* h    = ws + nA;
    float* agg1 = ws + nA + (size_t)n * 16;
    float* h2   = h;

    const int B = 256;
    auto blocks = [](long long work, int b) { return (int)((work + b - 1) / b); };

    k_init_deg  <<<blocks(n, B), B, 0, stream>>>(dis, n);
    k_deg_accum <<<blocks(e, B), B, 0, stream>>>(dst, dis, e);
    k_deg_to_dis<<<blocks(n, B), B, 0, stream>>>(dis, n);

    const int waves = (n + 15) / 16;              // one 16-row tile per wave
    const int wavesPerBlock = B / 32;
    k_gemm1_wmma<<<blocks(waves, wavesPerBlock), B, 0, stream>>>(x, W1, h, n);

    k_init_agg1<<<blocks((long long)n * 4, B), B, 0, stream>>>(h, dis, agg1, n);
    k_scatter1 <<<blocks((long long)e * 4, B), B, 0, stream>>>(src, dst, dis, h, agg1, e);

    k_layer2   <<<blocks(n, B), B, 0, stream>>>(agg1, b1, W2, h2, n);

    k_init_out <<<blocks(n, B), B, 0, stream>>>(h2, dis, b2, out, n);
    k_scatter2 <<<blocks(e, B), B, 0, stream>>>(src, dst, dis, h2, out, e);
}